// MedGCNLayer_45827301048843
// MI455X (gfx1250) — compile-verified
//
#include <hip/hip_runtime.h>
#include <hip/hip_bf16.h>

#define N_NODES 50000
#define N_EDGES 800000
#define N_REL   4
#define D_IN    128
#define D_OUT   128

typedef __attribute__((ext_vector_type(2))) float v2f;
typedef __attribute__((ext_vector_type(8))) float v8f;

// -------------------------------------------------------------------------
// Degree accumulation: deg[rows[e]] += vals[e]
// -------------------------------------------------------------------------
__global__ void medgcn_deg_kernel(const int* __restrict__ rows,
                                  const float* __restrict__ vals,
                                  float* __restrict__ deg) {
    int e = blockIdx.x * blockDim.x + threadIdx.x;
    if (e < N_EDGES) {
        atomicAdd(&deg[rows[e]], vals[e]);
    }
}

// -------------------------------------------------------------------------
// deg -> clip(deg, 1e-12)^-0.5
// -------------------------------------------------------------------------
__global__ void medgcn_invsqrt_kernel(float* __restrict__ deg) {
    int i = blockIdx.x * blockDim.x + threadIdx.x;
    if (i < N_NODES) {
        float d = deg[i];
        d = fmaxf(d, 1e-12f);
        deg[i] = rsqrtf(d);
    }
}

// -------------------------------------------------------------------------
// SpMM: one wave (32 lanes) per edge. Each lane handles 4 of the 128 feats.
//   AX[i] += (v * inv[i] * inv[j]) * X[j]
// float4 gather from X (global_load_b128) + 4x global_atomic_add_f32 scatter.
// X (25.6 MB) and AX (25.6 MB) are L2-resident (192 MB), so this is
// L2-bandwidth / atomic-throughput bound, which is the right regime.
// Edge count is an exact multiple of waves/block so EXEC stays uniform.
// -------------------------------------------------------------------------
__global__ void medgcn_spmm_kernel(const int* __restrict__ rows,
                                   const int* __restrict__ cols,
                                   const float* __restrict__ vals,
                                   const float* __restrict__ X,
                                   const float* __restrict__ inv,
                                   float* __restrict__ AX) {
    int tid  = blockIdx.x * blockDim.x + threadIdx.x;
    int edge = tid >> 5;          // wave per edge
    int lane = tid & 31;
    if (edge >= N_EDGES) return;

    int i = rows[edge];
    int j = cols[edge];
    float vn = vals[edge] * inv[i] * inv[j];

    const float4 xj = *(const float4*)(X + (size_t)j * D_IN + lane * 4);
    float* dst = AX + (size_t)i * D_IN + lane * 4;
    atomicAdd(dst + 0, vn * xj.x);
    atomicAdd(dst + 1, vn * xj.y);
    atomicAdd(dst + 2, vn * xj.z);
    atomicAdd(dst + 3, vn * xj.w);
}

// -------------------------------------------------------------------------
// GEMM via V_WMMA_F32_16X16X4_F32 (fp32 tensor path, wave32).
// One wave computes a 16x16 tile of out. M=50000 (=3125*16), K=N=128.
// A: (M x 128) row-major (X or AX). B: (128 x 128) row-major (W).
//   ACCUM=false: out = A@B + bias   (init pass, X @ W_self + b)
//   ACCUM=true : out += A@B         (accumulate pass, AX @ W_r)
// Compile-time mode -> straight-line epilogue (no per-element branches).
//
// Fragment layouts per CDNA5 ISA 7.12.2:
//   A 16x4:  lane&15 = M row; lane>>4 selects K pair {0,1} vs {2,3}
//            vgpr0 = A[M][k+koff], vgpr1 = A[M][k+koff+1]
//   B 4x16:  lane&15 = N col; same K-pair split
//   C/D:     vgpr g: N = lane&15, M = g + (lane>=16 ? 8 : 0)
// -------------------------------------------------------------------------
template <bool ACCUM>
__global__ void medgcn_gemm_wmma_kernel(const float* __restrict__ A,
                                        const float* __restrict__ B,
                                        const float* __restrict__ bias,
                                        float* __restrict__ out) {
    const int WAVES_PER_BLOCK = 4;
    int wave = blockIdx.x * WAVES_PER_BLOCK + (threadIdx.x >> 5);
    int lane = threadIdx.x & 31;

    const int TILES_N = D_OUT / 16;          // 8
    int tileM = wave / TILES_N;
    int tileN = wave % TILES_N;

    int mRow = tileM * 16 + (lane & 15);     // A row this lane supplies
    int nCol = tileN * 16 + (lane & 15);     // B col this lane supplies
    int kOff = (lane >> 4) * 2;              // 0 or 2

    const float* aPtr = A + (size_t)mRow * D_IN + kOff;
    const float* bPtr = B + (size_t)kOff * D_OUT + nCol;

    v8f c = {};
#pragma unroll 4
    for (int k = 0; k < D_IN; k += 4) {
        v2f a = *(const v2f*)(aPtr + k);                 // A[M][k+kOff], A[M][k+kOff+1]
        v2f b;
        b.x = bPtr[(size_t)k * D_OUT];                   // B[k+kOff][nCol]
        b.y = bPtr[(size_t)k * D_OUT + D_OUT];           // B[k+kOff+1][nCol]
        c = __builtin_amdgcn_wmma_f32_16x16x4_f32(
                /*neg_a=*/false, a, /*neg_b=*/false, b,
                /*c_mod=*/(short)0, c,
                /*reuse_a=*/false, /*reuse_b=*/false);
    }

    int rowBase = tileM * 16 + ((lane >> 4) * 8);
    float* o = out + (size_t)rowBase * D_OUT + nCol;

    if (ACCUM) {
#pragma unroll
        for (int g = 0; g < 8; ++g) {
            o[(size_t)g * D_OUT] += c[g];
        }
    } else {
        float badd = bias[nCol];
#pragma unroll
        for (int g = 0; g < 8; ++g) {
            o[(size_t)g * D_OUT] = c[g] + badd;
        }
    }
}

// -------------------------------------------------------------------------
// Launch
// -------------------------------------------------------------------------
extern "C" void kernel_launch(void* const* d_in, const int* in_sizes, int n_in,
                              void* d_out, int out_size, void* d_ws, size_t ws_size,
                              hipStream_t stream) {
    const float* X      = (const float*)d_in[0];
    const int*   rows   = (const int*)d_in[1];     // (N_REL, N_EDGES)
    const int*   cols   = (const int*)d_in[2];     // (N_REL, N_EDGES)
    const float* vals   = (const float*)d_in[3];   // (N_REL, N_EDGES)
    const float* W_rel  = (const float*)d_in[4];   // (N_REL, 128, 128)
    const float* W_self = (const float*)d_in[5];   // (128, 128)
    const float* bias   = (const float*)d_in[6];   // (128,)
    float*       out    = (float*)d_out;           // (N_NODES, 128)

    // workspace: AX (N*128 floats) then deg/inv (N floats)
    float* AX  = (float*)d_ws;
    float* deg = AX + (size_t)N_NODES * D_IN;
    const size_t scratchBytes = ((size_t)N_NODES * (D_IN + 1)) * sizeof(float);

    // GEMM grid: 3125*8 = 25000 tiles, 4 waves (128 thr) per block -> 6250 blocks
    const int GEMM_BLOCKS  = (N_NODES / 16) * (D_OUT / 16) / 4;
    const int GEMM_THREADS = 128;

    // 1) out = X @ W_self + bias
    medgcn_gemm_wmma_kernel<false><<<GEMM_BLOCKS, GEMM_THREADS, 0, stream>>>(
        X, W_self, bias, out);

    for (int r = 0; r < N_REL; ++r) {
        const int*   rr = rows + (size_t)r * N_EDGES;
        const int*   cc = cols + (size_t)r * N_EDGES;
        const float* vv = vals + (size_t)r * N_EDGES;

        // 2a) zero AX + deg (graph-capture-safe async memset)
        hipMemsetAsync(d_ws, 0, scratchBytes, stream);

        // 2b) deg[i] = sum of vals over edges with row i
        medgcn_deg_kernel<<<N_EDGES / 256, 256, 0, stream>>>(rr, vv, deg);

        // 2c) deg -> clip(deg,1e-12)^-0.5
        medgcn_invsqrt_kernel<<<(N_NODES + 255) / 256, 256, 0, stream>>>(deg);

        // 2d) AX = D^-1/2 A D^-1/2 @ X   (one wave per edge, 8 waves/block)
        medgcn_spmm_kernel<<<(N_EDGES * 32) / 256, 256, 0, stream>>>(
            rr, cc, vv, X, deg, AX);

        // 2e) out += AX @ W_rel[r]
        medgcn_gemm_wmma_kernel<true><<<GEMM_BLOCKS, GEMM_THREADS, 0, stream>>>(
            AX, W_rel + (size_t)r * D_IN * D_OUT, nullptr, out);
    }
}